// RaFC_32736240730244
// MI455X (gfx1250) — compile-verified
//
#include <hip/hip_runtime.h>
#include <hip/hip_bf16.h>
#include <stdint.h>

typedef __attribute__((ext_vector_type(2))) float v2f;
typedef __attribute__((ext_vector_type(8))) float v8f;

#define TILE_W 128
#define TILE_H 16
#define HALO   2
#define IN_W   (TILE_W + 2*HALO)   // 132
#define IN_H   (TILE_H + 2*HALO)   // 20
#define LSTR   136                 // pad: (2*136)%64==16 -> lane halves hit disjoint banks
#define IMG    256
#define PLANE  (IMG*IMG)
#define NLOAD  11                  // ceil(20*132/256)
#define CPB    8                   // channels per block (amortize gamma/tanh/weights)

// Normalized 1D Gaussian weights (reference 2D kernel is separable: outer(g1n,g1n))
#define G5_0 0.15246915f
#define G5_1 0.22184131f
#define G5_2 0.25137913f
#define G3_0 0.27406862f
#define G3_1 0.45186276f

__device__ __forceinline__ float w5of(int t) {
    float w = 0.0f;
    w = (t == 0) ? G5_0 : w;
    w = (t == 1) ? G5_1 : w;
    w = (t == 2) ? G5_2 : w;
    w = (t == 3) ? G5_1 : w;
    w = (t == 4) ? G5_0 : w;
    return w;
}
__device__ __forceinline__ float w3of(int t) {
    float w = 0.0f;
    w = (t == 1) ? G3_0 : w;
    w = (t == 2) ? G3_1 : w;
    w = (t == 3) ? G3_0 : w;
    return w;
}

// Issue one tile's worth of async global->LDS loads (exactly NLOAD per wave,
// deterministic ASYNCcnt bookkeeping for the software pipeline).
__device__ __forceinline__ void issue_async_tile(uint64_t base, uint32_t ldsbase,
                                                 const uint32_t (&goff)[NLOAD],
                                                 const uint32_t (&loff)[NLOAD]) {
    #pragma unroll
    for (int it = 0; it < NLOAD; ++it) {
        uint32_t lds = ldsbase + loff[it];
        asm volatile("global_load_async_to_lds_b32 %0, %1, %2"
                     :: "v"(lds), "v"(goff[it]), "s"(base) : "memory");
    }
}

__global__ __launch_bounds__(256) void scale_space_fused(
    const float* __restrict__ x, const float* __restrict__ gamma,
    float* __restrict__ val_out, float* __restrict__ conf_out)
{
    __shared__ float s_in0[IN_H * LSTR];
    __shared__ float s_in1[IN_H * LSTR];
    __shared__ float s_h5[IN_H * LSTR];
    __shared__ float s_h3[IN_H * LSTR];

    const int tid   = threadIdx.x;
    const int zb    = blockIdx.z;          // b*4 + channel-group
    const int bidx  = zb >> 2;
    const int ch0   = (zb & 3) * CPB;
    const int row0  = blockIdx.y * TILE_H;
    const int col0g = blockIdx.x * TILE_W;

    // ================= per-block invariant precompute =================
    // (a) async-load addressing: replicate-clamped halo tile
    uint32_t goff[NLOAD], loff[NLOAD];
    #pragma unroll
    for (int it = 0; it < NLOAD; ++it) {
        int e = min(it * 256 + tid, IN_H * IN_W - 1);  // tail lanes duplicate last elem
        int r = e / IN_W;
        int c = e - r * IN_W;
        int gr = min(max(row0  - HALO + r, 0), IMG - 1);
        int gc = min(max(col0g - HALO + c, 0), IMG - 1);
        goff[it] = (uint32_t)(((gr << 8) + gc) << 2);
        loff[it] = (uint32_t)((r * LSTR + c) << 2);
    }
    const uint32_t in0base = (uint32_t)(uintptr_t)s_in0;
    const uint32_t in1base = (uint32_t)(uintptr_t)s_in1;

    // (b) horizontal-pass element offsets (same layout for s_in/s_h5/s_h3)
    uint32_t eoff[10];
    #pragma unroll
    for (int it = 0; it < 10; ++it) {
        int e = it * 256 + tid;            // 0..2559
        eoff[it] = (uint32_t)((e >> 7) * LSTR + (e & 127));
    }

    // (c) WMMA lane mapping + banded A-matrix weights (constant per lane)
    const int lane  = tid & 31;
    const int wv    = tid >> 5;
    const int colw  = wv * 16;
    const int nIdx  = lane & 15;           // B/C/D column == A row
    const int half  = (lane >> 4) << 1;    // A/B VGPR K-split: +2 for lanes 16..31
    const int mbase = (lane >> 4) * 8;     // C/D rows: lanes 16..31 hold M=8..15
    float a5x[5], a5y[5], a3x[5], a3y[5];
    #pragma unroll
    for (int kk = 0; kk < 5; ++kk) {
        int j0 = 4 * kk + half, j1 = j0 + 1;
        a5x[kk] = w5of(j0 - nIdx); a5y[kk] = w5of(j1 - nIdx);
        a3x[kk] = w3of(j0 - nIdx); a3y[kk] = w3of(j1 - nIdx);
    }

    // (d) gamma-derived blend state, shared by all CPB channels; conf written once
    const int gcol = col0g + colw + nIdx;
    float alpha[8];
    int   lcode[8], ucode[8];              // 0 -> blur5, 1 -> blur3, 2 -> identity
    {
        const float* gplane = gamma + (size_t)bidx * PLANE;
        float* conf = conf_out + (size_t)bidx * PLANE;
        #pragma unroll
        for (int i = 0; i < 8; ++i) {
            int grow = row0 + mbase + i;
            float g = gplane[(grow << 8) + gcol];
            float s = tanhf(g);
            float fidx = (s + 1.0f) * 2.0f;            // in [0,4]
            float fl = floorf(fidx);
            int lb = min(max((int)fl, 0), 4);
            int ub = min(lb + 1, 4);
            alpha[i] = fidx - fl;
            lcode[i] = (lb == 2) ? 2 : (lb & 1);
            ucode[i] = (ub == 2) ? 2 : (ub & 1);
            if (ch0 == 0)
                conf[(grow << 8) + gcol] = (1.0f - fabsf(s)) * 2.0f;
        }
    }

    // ================= channel loop, double-buffered async pipeline =================
    const float* xb = x       + (size_t)(bidx * 32 + ch0) * PLANE;
    float*       vb = val_out + (size_t)(bidx * 32 + ch0) * PLANE;

    issue_async_tile((uint64_t)(uintptr_t)xb, in0base, goff, loff);   // prefetch ch 0

    for (int c = 0; c < CPB; ++c) {
        const int cur = c & 1;
        if (c + 1 < CPB)   // prefetch next channel into the other buffer
            issue_async_tile((uint64_t)(uintptr_t)(xb + (size_t)(c + 1) * PLANE),
                             cur ? in0base : in1base, goff, loff);
        // current channel's NLOAD loads are the oldest in flight (in-order completion)
        if (c + 1 < CPB) asm volatile("s_wait_asynccnt 0xb" ::: "memory");
        else             asm volatile("s_wait_asynccnt 0x0" ::: "memory");
        __syncthreads();

        const float* sin = cur ? s_in1 : s_in0;

        // horizontal blur5 / blur3 for all 20 rows
        #pragma unroll
        for (int it = 0; it < 10; ++it) {
            const float* row = sin + eoff[it];
            float v0 = row[0], v1 = row[1], v2 = row[2], v3 = row[3], v4 = row[4];
            s_h5[eoff[it]] = G5_0 * (v0 + v4) + G5_1 * (v1 + v3) + G5_2 * v2;
            s_h3[eoff[it]] = G3_0 * (v1 + v3) + G3_1 * v2;
        }
        __syncthreads();

        // vertical blur: chained V_WMMA_F32_16X16X4_F32, K = 20 (16 + halo)
        v8f c5 = {0.f,0.f,0.f,0.f,0.f,0.f,0.f,0.f};
        v8f c3 = {0.f,0.f,0.f,0.f,0.f,0.f,0.f,0.f};
        #pragma unroll
        for (int kk = 0; kk < 5; ++kk) {
            const int j0 = 4 * kk + half, j1 = j0 + 1;
            v2f a5, b5, a3, b3;
            a5.x = a5x[kk]; a5.y = a5y[kk];
            a3.x = a3x[kk]; a3.y = a3y[kk];
            b5.x = s_h5[j0 * LSTR + colw + nIdx];
            b5.y = s_h5[j1 * LSTR + colw + nIdx];
            b3.x = s_h3[j0 * LSTR + colw + nIdx];
            b3.y = s_h3[j1 * LSTR + colw + nIdx];
            c5 = __builtin_amdgcn_wmma_f32_16x16x4_f32(false, a5, false, b5,
                                                       (short)0, c5, false, false);
            c3 = __builtin_amdgcn_wmma_f32_16x16x4_f32(false, a3, false, b3,
                                                       (short)0, c3, false, false);
        }

        // blend with precomputed per-pixel state and store
        float* vo = vb + (size_t)c * PLANE;
        #pragma unroll
        for (int i = 0; i < 8; ++i) {
            float xc  = sin[(mbase + i + HALO) * LSTR + (colw + nIdx + HALO)];
            float b5v = c5[i];
            float b3v = c3[i];
            float lv = (lcode[i] == 2) ? xc : ((lcode[i] == 1) ? b3v : b5v);
            float uv = (ucode[i] == 2) ? xc : ((ucode[i] == 1) ? b3v : b5v);
            vo[((row0 + mbase + i) << 8) + gcol] = (1.0f - alpha[i]) * lv + alpha[i] * uv;
        }
        __syncthreads();   // all reads of sin/s_h* done before buffers are overwritten
    }
}

extern "C" void kernel_launch(void* const* d_in, const int* in_sizes, int n_in,
                              void* d_out, int out_size, void* d_ws, size_t ws_size,
                              hipStream_t stream) {
    const float* x     = (const float*)d_in[0];   // [8,32,256,256]
    const float* gamma = (const float*)d_in[1];   // [8,1,256,256]
    float* val  = (float*)d_out;                                // 8*32*256*256
    float* conf = (float*)d_out + (size_t)8 * 32 * PLANE;       // 8*1*256*256
    dim3 grid(IMG / TILE_W, IMG / TILE_H, 8 * (32 / CPB));      // (2, 16, 32)
    scale_space_fused<<<grid, 256, 0, stream>>>(x, gamma, val, conf);
}